// ManifoldAgentSystem_9921374454474
// MI455X (gfx1250) — compile-verified
//
#include <hip/hip_runtime.h>

typedef __attribute__((ext_vector_type(2))) float v2f;
typedef __attribute__((ext_vector_type(8))) float v8f;

#define NAG 16
#define GRD 4096
#define AB_K 80   // GEMM K: 64 (tr) + 8 (mu terms) + 2 (consts) + 6 pad

// In-place Gauss-Jordan inverse of 8x8 (no pivoting: matrices are I + small
// perturbation or SPD). Returns log|det|.
__device__ __forceinline__ float invert8_inplace(float M[8][8]) {
  float ld = 0.f;
#pragma unroll
  for (int k = 0; k < 8; ++k) {
    float p = M[k][k];
    ld += __logf(fabsf(p));
    float pinv = 1.0f / p;
    M[k][k] = pinv;
#pragma unroll
    for (int j = 0; j < 8; ++j)
      if (j != k) M[k][j] *= pinv;
#pragma unroll
    for (int i = 0; i < 8; ++i) {
      if (i == k) continue;
      float f = M[i][k];
      M[i][k] = -f * pinv;
#pragma unroll
      for (int j = 0; j < 8; ++j)
        if (j != k) M[i][j] -= f * M[k][j];
    }
  }
  return ld;
}

// log|det| only (destroys M).
__device__ __forceinline__ float logdet8_inplace(float M[8][8]) {
  float ld = 0.f;
#pragma unroll
  for (int k = 0; k < 8; ++k) {
    float p = M[k][k];
    ld += __logf(fabsf(p));
    float pinv = 1.0f / p;
#pragma unroll
    for (int i = k + 1; i < 8; ++i) {
      float f = M[i][k] * pinv;
#pragma unroll
      for (int j = k + 1; j < 8; ++j) M[i][j] -= f * M[k][j];
    }
  }
  return ld;
}

// Phase 1: per (n,g) build GEMM rows.
//   A row: [ P + m m^T (64) | m (8) | 2 ld|T| - ldS (1) | 1 (1) | 0 pad ]
//   B row: [ Q (64) | -2 r (8) | 1 (1) | s + ldS - 2 ld|omega| - K (1) | 0 pad ]
__global__ void __launch_bounds__(256)
manifold_phase1(const float* __restrict__ omega, const float* __restrict__ links,
                const float* __restrict__ mu_q, const float* __restrict__ sigma_q,
                float* __restrict__ Aws, float* __restrict__ Bws) {
  int t = blockIdx.x * blockDim.x + threadIdx.x;   // 0 .. N*G-1
  if (t >= NAG * GRD) return;
  int n = t >> 12;            // t / GRD
  int g = t & (GRD - 1);      // t % GRD

  const float* om_p = omega   + (size_t)t * 64;
  const float* l0_p = links   + ((size_t)(n * 2 + 0) * GRD + g) * 64;
  const float* l1_p = links   + ((size_t)(n * 2 + 1) * GRD + g) * 64;
  const float* mu_p = mu_q    + (size_t)t * 8;
  const float* sg_p = sigma_q + (size_t)t * 64;

  float Om[8][8], M[8][8], S[8][8], mu[8], m[8], tv[8], u[8];

#pragma unroll
  for (int a = 0; a < 8; ++a)
#pragma unroll
    for (int b = 0; b < 8; ++b) {
      int idx = a * 8 + b;
      Om[a][b] = om_p[idx];
      // V = I + sum_d (link_d - I) = l0 + l1 - I
      M[a][b] = l0_p[idx] + l1_p[idx] - ((a == b) ? 1.0f : 0.0f);
    }

  // T = Om @ V, column by column in place (col c of T only needs col c of V)
#pragma unroll
  for (int c = 0; c < 8; ++c) {
#pragma unroll
    for (int a = 0; a < 8; ++a) {
      float s = 0.f;
#pragma unroll
      for (int b = 0; b < 8; ++b) s += Om[a][b] * M[b][c];
      tv[a] = s;
    }
#pragma unroll
    for (int a = 0; a < 8; ++a) M[a][c] = tv[a];
  }

  float ldT = invert8_inplace(M);        // M = W = T^{-1}

#pragma unroll
  for (int a = 0; a < 8; ++a) mu[a] = mu_p[a];
#pragma unroll
  for (int a = 0; a < 8; ++a) {
    float s = 0.f;
#pragma unroll
    for (int b = 0; b < 8; ++b) s += M[a][b] * mu[b];
    m[a] = s;                            // m = W mu
  }
#pragma unroll
  for (int a = 0; a < 8; ++a)
#pragma unroll
    for (int b = 0; b < 8; ++b) S[a][b] = sg_p[a * 8 + b];

  float* Ar = Aws + ((size_t)g * NAG + n) * AB_K;
  float* Br = Bws + ((size_t)g * NAG + n) * AB_K;

  // A[0:64] = W Sigma W^T + m m^T   (streamed row-wise)
#pragma unroll
  for (int a = 0; a < 8; ++a) {
#pragma unroll
    for (int d = 0; d < 8; ++d) {
      float s = 0.f;
#pragma unroll
      for (int c = 0; c < 8; ++c) s += M[a][c] * S[c][d];
      tv[d] = s;
    }
#pragma unroll
    for (int b = 0; b < 8; ++b) {
      float s = m[a] * m[b];
#pragma unroll
      for (int d = 0; d < 8; ++d) s += tv[d] * M[b][d];
      Ar[a * 8 + b] = s;
    }
  }
#pragma unroll
  for (int a = 0; a < 8; ++a) Ar[64 + a] = m[a];

  float ldS = invert8_inplace(S);        // S = Sigma^{-1} (SPD)

  // u = Sinv mu, s = mu.u, r = Om^T u
#pragma unroll
  for (int a = 0; a < 8; ++a) {
    float s = 0.f;
#pragma unroll
    for (int b = 0; b < 8; ++b) s += S[a][b] * mu[b];
    u[a] = s;
  }
  float sscal = 0.f;
#pragma unroll
  for (int a = 0; a < 8; ++a) sscal += u[a] * mu[a];
#pragma unroll
  for (int a = 0; a < 8; ++a) {
    float s = 0.f;
#pragma unroll
    for (int c = 0; c < 8; ++c) s += Om[c][a] * u[c];
    Br[64 + a] = -2.0f * s;
  }

  // B[0:64] = Q = Om^T Sinv Om   (streamed row-wise)
#pragma unroll
  for (int a = 0; a < 8; ++a) {
#pragma unroll
    for (int d = 0; d < 8; ++d) {
      float s = 0.f;
#pragma unroll
      for (int c = 0; c < 8; ++c) s += Om[c][a] * S[c][d];
      tv[d] = s;
    }
#pragma unroll
    for (int b = 0; b < 8; ++b) {
      float s = 0.f;
#pragma unroll
      for (int d = 0; d < 8; ++d) s += tv[d] * Om[d][b];
      Br[a * 8 + b] = s;
    }
  }

  float ldO = logdet8_inplace(Om);       // Om destroyed (no longer needed)

  Ar[72] = 2.0f * ldT - ldS;
  Ar[73] = 1.0f;
  Br[72] = 1.0f;
  Br[73] = sscal + ldS - 2.0f * ldO - 8.0f;
#pragma unroll
  for (int k = 74; k < AB_K; ++k) { Ar[k] = 0.0f; Br[k] = 0.0f; }
}

// Phase 2: per g, E[i,j] = 0.5 * (A_i . B_j) via 20 chained f32 WMMAs.
// 8 waves per block, one wave per g; LDS transpose for coalesced stores.
__global__ void __launch_bounds__(256)
manifold_phase2(const float* __restrict__ Aws, const float* __restrict__ Bws,
                float* __restrict__ out) {
  __shared__ float tile[8][256];
  const int lane = threadIdx.x & 31;
  const int wave = threadIdx.x >> 5;           // wave32
  const int g    = blockIdx.x * 8 + wave;
  const int row  = lane & 15;                  // M for A-frag, N for B-frag
  const int hi   = lane >> 4;                  // upper half-wave holds K+2,K+3

  const float* Ar = Aws + ((size_t)g * NAG + row) * AB_K + 2 * hi;
  const float* Br = Bws + ((size_t)g * NAG + row) * AB_K + 2 * hi;

  v8f acc = {0.f, 0.f, 0.f, 0.f, 0.f, 0.f, 0.f, 0.f};
#pragma unroll
  for (int t = 0; t < AB_K / 4; ++t) {
    v2f a = *(const v2f*)(Ar + 4 * t);         // A frag: 16x4 f32, 2 VGPRs
    v2f b = *(const v2f*)(Br + 4 * t);         // B frag: 4x16 f32, 2 VGPRs
    acc = __builtin_amdgcn_wmma_f32_16x16x4_f32(
        /*neg_a=*/false, a, /*neg_b=*/false, b,
        /*c_mod=*/(short)0, acc, /*reuse_a=*/false, /*reuse_b=*/false);
  }

  // C/D layout: VGPR v, lanes 0-15 -> (M=v, N=lane); lanes 16-31 -> (M=v+8)
#pragma unroll
  for (int v = 0; v < 8; ++v) {
    int i = v + 8 * hi;
    int j = row;
    float val = 0.5f * acc[v];
    if (i == j) val = 0.0f;                    // zero diagonal
    tile[wave][i * 16 + j] = val;
  }
  __syncthreads();

  // coalesced store: thread p owns pair (i,j)=p, writes 8 consecutive g
  int p = threadIdx.x;
  size_t base = (size_t)p * GRD + (size_t)blockIdx.x * 8;
  float4 v0 = make_float4(tile[0][p], tile[1][p], tile[2][p], tile[3][p]);
  float4 v1 = make_float4(tile[4][p], tile[5][p], tile[6][p], tile[7][p]);
  *(float4*)(out + base)     = v0;
  *(float4*)(out + base + 4) = v1;
}

extern "C" void kernel_launch(void* const* d_in, const int* in_sizes, int n_in,
                              void* d_out, int out_size, void* d_ws, size_t ws_size,
                              hipStream_t stream) {
  (void)in_sizes; (void)n_in; (void)out_size; (void)ws_size;
  const float* omega = (const float*)d_in[0];
  const float* links = (const float*)d_in[1];
  const float* mu_q  = (const float*)d_in[2];
  const float* sigma = (const float*)d_in[3];
  float* out = (float*)d_out;

  float* Aws = (float*)d_ws;                       // G*16*80 floats = ~21 MB
  float* Bws = Aws + (size_t)GRD * NAG * AB_K;     // same again

  manifold_phase1<<<(NAG * GRD) / 256, 256, 0, stream>>>(omega, links, mu_q,
                                                         sigma, Aws, Bws);
  manifold_phase2<<<GRD / 8, 256, 0, stream>>>(Aws, Bws, out);
}